// GINWrapper_85624468013528
// MI455X (gfx1250) — compile-verified
//
#include <hip/hip_runtime.h>

typedef __attribute__((ext_vector_type(2))) float v2f;
typedef __attribute__((ext_vector_type(8))) float v8f;
typedef __attribute__((ext_vector_type(4))) unsigned int v4u;
typedef __attribute__((ext_vector_type(8))) int v8i;
typedef __attribute__((ext_vector_type(4))) int v4i;

#define N_NODES 20000
#define E_EDGES 320000
#define DIM     256
#define LAYERS  4
#define GRAPHS  128
#define OUTD    64
#define ROWS_PER_BLOCK 32
#define LDS_STRIDE 260   // 256 + 4 pad floats per row (matches TDM pad config)

#if defined(__has_builtin)
#  if __has_builtin(__builtin_amdgcn_tensor_load_to_lds)
#    define USE_TDM 1
#  endif
#endif
#ifndef USE_TDM
#  define USE_TDM 0
#endif

// ---------------------------------------------------------------------------
// float4 copy (z = h, and x -> h)
// ---------------------------------------------------------------------------
__global__ __launch_bounds__(256) void copy_f4(const float4* __restrict__ src,
                                               float4* __restrict__ dst, int n4) {
    int i = blockIdx.x * blockDim.x + threadIdx.x;
    if (i < n4) dst[i] = src[i];
}

__global__ __launch_bounds__(256) void zero_f(float* __restrict__ p, int n) {
    int i = blockIdx.x * blockDim.x + threadIdx.x;
    if (i < n) p[i] = 0.0f;
}

// ---------------------------------------------------------------------------
// Edge scatter: z[dst] += h[src]; 64 threads (float4 each) per edge.
// h/z are L2-resident (20.5 MB each, 192 MB L2) -> atomics at L2 throughput.
// ---------------------------------------------------------------------------
__global__ __launch_bounds__(256)
void scatter_add(const float* __restrict__ h, const int* __restrict__ esrc,
                 const int* __restrict__ edst, float* __restrict__ z) {
    int i = blockIdx.x * blockDim.x + threadIdx.x;
    if (i >= E_EDGES * (DIM / 4)) return;
    int e = i >> 6;             // edge id
    int c = (i & 63) << 2;      // feature chunk
    int s = esrc[e];
    int d = edst[e];
    float4 v = *(const float4*)(h + (size_t)s * DIM + c);
    float* zp = z + (size_t)d * DIM + c;
    atomicAdd(zp + 0, v.x);
    atomicAdd(zp + 1, v.y);
    atomicAdd(zp + 2, v.z);
    atomicAdd(zp + 3, v.w);
}

// ---------------------------------------------------------------------------
// Fused GEMM + bias + optional ReLU:
//   Out[M x 256] = act(A[M x 256] @ W[256 x 256] + b)
// Wave computes a 32x16 output (two stacked 16x16 WMMA tiles) so each B
// fragment is reused by 2 WMMAs. Block = 8 waves = 32 rows x 128 cols.
// A panel (32x256) is staged to LDS by the Tensor Data Mover with LDS
// padding (row stride 260 floats -> conflict-free ds_load_b64 fragments).
// ---------------------------------------------------------------------------
__global__ __launch_bounds__(256)
void gin_gemm_wmma(const float* __restrict__ A, const float* __restrict__ W,
                   const float* __restrict__ bias, float* __restrict__ Out,
                   int relu) {
    __shared__ float sA[ROWS_PER_BLOCK * LDS_STRIDE];

    const int tid   = threadIdx.x;
    const int wave  = tid >> 5;
    const int lane  = tid & 31;
    const int panel = blockIdx.x >> 1;                          // 0..624
    const int row0  = panel * ROWS_PER_BLOCK;
    const int col0  = ((((blockIdx.x & 1) << 3) + wave) << 4);  // 0..240

#if USE_TDM
    if (wave == 0) {
        // ---- Tensor DMA descriptor: 2D tile [tile_dim1=32 rows x tile_dim0=256],
        //      data_size=4B, tensor_dim0_stride=256, pad 4 DWORDs per 256 DWORDs.
        unsigned long long ga = (unsigned long long)(A + (size_t)row0 * DIM);
        unsigned int lb = (unsigned int)(unsigned long long)&sA[0];
        v4u g0;
        g0.x = 1u;                                    // count=1 (valid user D#)
        g0.y = lb;                                    // lds_addr
        g0.z = (unsigned int)ga;                      // global_addr[31:0]
        g0.w = (unsigned int)((ga >> 32) & 0x01FFFFFFu) | (2u << 30); // addr hi | type=2
        v8i g1;
        g1[0] = (int)((2u << 16)                      // data_size = 4B
                    | (1u << 20)                      // pad_enable
                    | (7u << 22)                      // pad_interval: every 256 DWORDs
                    | (3u << 25));                    // pad_amount: 4 DWORDs
        g1[1] = (int)(((unsigned)DIM & 0xFFFFu) << 16);        // tensor_dim0[15:0]
        g1[2] = (int)(((unsigned)ROWS_PER_BLOCK) << 16);       // tensor_dim1[15:0]
        g1[3] = (int)(((unsigned)DIM) << 16);                  // tile_dim0 = 256
        g1[4] = ROWS_PER_BLOCK;                                // tile_dim1=32, tile_dim2=0
        g1[5] = DIM;                                           // tensor_dim0_stride
        g1[6] = 0;
        g1[7] = 0;
        v4i gz = {0, 0, 0, 0};                                 // groups 2/3 unused (2D)
#if defined(__clang_major__) && (__clang_major__ >= 23)
        v8i z8 = {0, 0, 0, 0, 0, 0, 0, 0};
        __builtin_amdgcn_tensor_load_to_lds(g0, g1, gz, gz, z8, 0);
#else
        __builtin_amdgcn_tensor_load_to_lds(g0, g1, gz, gz, 0);
#endif
        __builtin_amdgcn_s_wait_tensorcnt(0);
    }
    __syncthreads();
#else
    for (int idx = tid; idx < ROWS_PER_BLOCK * DIM; idx += 256) {
        int r = idx >> 8;
        int c = idx & 255;
        sA[r * LDS_STRIDE + c] = A[(size_t)(row0 + r) * DIM + c];
    }
    __syncthreads();
#endif

    const int hi   = lane >> 4;   // half-wave selects K pair {0,1} vs {2,3}
    const int lidx = lane & 15;   // A: row m, B: col n

    const float* aRow0 = &sA[lidx * LDS_STRIDE + 2 * hi];          // rows 0..15
    const float* aRow1 = &sA[(lidx + 16) * LDS_STRIDE + 2 * hi];   // rows 16..31
    const float* bCol  = W + (size_t)(2 * hi) * DIM + col0 + lidx;

    v8f acc0 = {};
    v8f acc1 = {};
#pragma unroll 4
    for (int k = 0; k < DIM; k += 4) {
        v2f b;
        b.x = bCol[(size_t)k * DIM];                 // W[k+2hi][n]
        b.y = bCol[(size_t)k * DIM + DIM];           // W[k+2hi+1][n]
        v2f a0 = *(const v2f*)(aRow0 + k);
        v2f a1 = *(const v2f*)(aRow1 + k);
        acc0 = __builtin_amdgcn_wmma_f32_16x16x4_f32(
            false, a0, false, b, (short)0, acc0, false, false);
        acc1 = __builtin_amdgcn_wmma_f32_16x16x4_f32(
            false, a1, false, b, (short)0, acc1, false, false);
    }

    // C/D layout: lane holds N = lidx, VGPR v -> M = v + 8*hi
    float bv = bias[col0 + lidx];
    float* orow0 = Out + (size_t)row0 * DIM + col0 + lidx;
    float* orow1 = orow0 + (size_t)16 * DIM;
#pragma unroll
    for (int v = 0; v < 8; ++v) {
        float v0 = acc0[v] + bv;
        float v1 = acc1[v] + bv;
        if (relu) { v0 = fmaxf(v0, 0.0f); v1 = fmaxf(v1, 0.0f); }
        orow0[(size_t)(v + 8 * hi) * DIM] = v0;
        orow1[(size_t)(v + 8 * hi) * DIM] = v1;
    }
}

// ---------------------------------------------------------------------------
// Global add pool: pooled[batch_ids[n]] += h[n]; 64 threads (float4) per node.
// ---------------------------------------------------------------------------
__global__ __launch_bounds__(256)
void pool_add(const float* __restrict__ h, const int* __restrict__ bid,
              float* __restrict__ pooled) {
    int i = blockIdx.x * blockDim.x + threadIdx.x;
    if (i >= N_NODES * (DIM / 4)) return;
    int n = i >> 6;
    int c = (i & 63) << 2;
    int g = bid[n];
    float4 v = *(const float4*)(h + (size_t)n * DIM + c);
    float* pp = pooled + (size_t)g * DIM + c;
    atomicAdd(pp + 0, v.x);
    atomicAdd(pp + 1, v.y);
    atomicAdd(pp + 2, v.z);
    atomicAdd(pp + 3, v.w);
}

// ---------------------------------------------------------------------------
// Tiny output projection: out[128 x 64] = pooled[128 x 256] @ Wout + bout
// ---------------------------------------------------------------------------
__global__ __launch_bounds__(256)
void out_gemm(const float* __restrict__ pooled, const float* __restrict__ Wout,
              const float* __restrict__ bout, float* __restrict__ out) {
    int i = blockIdx.x * blockDim.x + threadIdx.x;
    if (i >= GRAPHS * OUTD) return;
    int g = i / OUTD;
    int o = i % OUTD;
    float acc = bout[o];
    const float* p = pooled + (size_t)g * DIM;
#pragma unroll 8
    for (int d = 0; d < DIM; ++d) acc = fmaf(p[d], Wout[d * OUTD + o], acc);
    out[i] = acc;
}

// ---------------------------------------------------------------------------
extern "C" void kernel_launch(void* const* d_in, const int* in_sizes, int n_in,
                              void* d_out, int out_size, void* d_ws, size_t ws_size,
                              hipStream_t stream) {
    (void)in_sizes; (void)n_in; (void)out_size; (void)ws_size;

    const float* x    = (const float*)d_in[0];
    const int*   esrc = (const int*)d_in[1];
    const int*   edst = (const int*)d_in[2];
    const int*   bid  = (const int*)d_in[3];
    const float* W1   = (const float*)d_in[4];
    const float* b1   = (const float*)d_in[5];
    const float* W2   = (const float*)d_in[6];
    const float* b2   = (const float*)d_in[7];
    const float* Wout = (const float*)d_in[8];
    const float* bout = (const float*)d_in[9];

    float* out    = (float*)d_out;                 // [128 x 64]
    float* pooled = out + GRAPHS * OUTD;           // [128 x 256] (also an output)

    const size_t ND = (size_t)N_NODES * DIM;
    float* h = (float*)d_ws;       // current features
    float* z = h + ND;             // h + aggregate
    float* t = z + ND;             // hidden after first linear

    const int ND4 = N_NODES * DIM / 4;             // 1,280,000 float4
    const int cpBlocks = (ND4 + 255) / 256;
    const int scBlocks = (E_EDGES * (DIM / 4) + 255) / 256;
    const int gemmBlocks = (N_NODES / ROWS_PER_BLOCK) * 2;   // 1250
    const int plBlocks = (N_NODES * (DIM / 4) + 255) / 256;

    copy_f4<<<cpBlocks, 256, 0, stream>>>((const float4*)x, (float4*)h, ND4);

    for (int l = 0; l < LAYERS; ++l) {
        copy_f4<<<cpBlocks, 256, 0, stream>>>((const float4*)h, (float4*)z, ND4);
        scatter_add<<<scBlocks, 256, 0, stream>>>(h, esrc, edst, z);
        gin_gemm_wmma<<<gemmBlocks, 256, 0, stream>>>(
            z, W1 + (size_t)l * DIM * DIM, b1 + (size_t)l * DIM, t, 1);
        gin_gemm_wmma<<<gemmBlocks, 256, 0, stream>>>(
            t, W2 + (size_t)l * DIM * DIM, b2 + (size_t)l * DIM, h,
            (l < LAYERS - 1) ? 1 : 0);
    }

    zero_f<<<(GRAPHS * (OUTD + DIM) + 255) / 256, 256, 0, stream>>>(
        out, GRAPHS * (OUTD + DIM));
    pool_add<<<plBlocks, 256, 0, stream>>>(h, bid, pooled);
    out_gemm<<<(GRAPHS * OUTD + 255) / 256, 256, 0, stream>>>(pooled, Wout, bout, out);
}